// DiffNet_Encoder_78683800863294
// MI455X (gfx1250) — compile-verified
//
#include <hip/hip_runtime.h>

#define USER_NUM 200000
#define ITEM_NUM 100000
#define EMB 64
#define NE 3200000

typedef __attribute__((ext_vector_type(2))) float v2f;
typedef __attribute__((ext_vector_type(8))) float v8f;

// ---------------- zero / copy helpers ----------------
__global__ void __launch_bounds__(256) zero_kernel(float* __restrict__ p, int n) {
  int i = blockIdx.x * blockDim.x + threadIdx.x;
  if (i < n) p[i] = 0.0f;
}

__global__ void __launch_bounds__(256) copy4_kernel(const float4* __restrict__ src,
                                                    float4* __restrict__ dst, int n4) {
  int i = blockIdx.x * blockDim.x + threadIdx.x;
  if (i < n4) dst[i] = src[i];
}

// ---------------- edge-parallel COO SpMM with f32 global atomics ----------------
// One wave (32 lanes) per edge; each lane handles 2 consecutive floats of the
// 64-wide embedding row. Gather source fits in 192MB L2 -> atomics & gathers
// stay on-chip.
__global__ void __launch_bounds__(256) spmm_atomic_kernel(
    const int* __restrict__ rows, const int* __restrict__ cols,
    const float* __restrict__ vals, const float* __restrict__ X,
    float* __restrict__ out, int ne) {
  int warp = (blockIdx.x * blockDim.x + threadIdx.x) >> 5;
  int lane = threadIdx.x & 31;
  if (warp >= ne) return;
  int r = rows[warp];
  int c = cols[warp];
  float v = vals[warp];
  const float2* x2 = (const float2*)(X + (size_t)c * EMB);
  float* o = out + (size_t)r * EMB;
  float2 x = x2[lane];
  unsafeAtomicAdd(&o[2 * lane + 0], v * x.x);
  unsafeAtomicAdd(&o[2 * lane + 1], v * x.y);
}

// ---------------- dense layer: Uout = relu([agg | U] @ W^T + b) via FP32 WMMA ----
// W is [64 out, 128 in] row-major (torch Linear layout).
// Wave-level tile: M=16 rows x N=64 outputs, K=128 in steps of 4
// (V_WMMA_F32_16X16X4_F32). One block = 8 waves = 128 rows.
__global__ void __launch_bounds__(256) dense_layer_kernel(
    const float* __restrict__ agg, const float* __restrict__ U,
    const float* __restrict__ W, const float* __restrict__ bias,
    float* __restrict__ Uout) {
  const int wave = threadIdx.x >> 5;
  const int lane = threadIdx.x & 31;
  const int m_base = (blockIdx.x * 8 + wave) * 16;
  if (m_base >= USER_NUM) return;  // wave-uniform exit; EXEC stays all-1 for WMMA

  const int mrow = m_base + (lane & 15);
  const int koff = (lane >> 4) << 1;  // lanes 0-15: K {0,1}; lanes 16-31: K {2,3}
  const int ncol = lane & 15;

  v8f acc0 = {}, acc1 = {}, acc2 = {}, acc3 = {};

  // half 0: K = 0..63 comes from agg; half 1: K = 64..127 comes from U
#pragma unroll
  for (int half = 0; half < 2; ++half) {
    const float* __restrict__ src = half ? U : agg;
    const float* hb = src + (size_t)mrow * EMB;
#pragma unroll
    for (int kk = 0; kk < 16; ++kk) {
      const int k0 = kk * 4;            // column within src half
      const int kg = half * 64 + k0;    // column within W row (K dim)
      v2f a;
      a.x = hb[k0 + koff];
      a.y = hb[k0 + koff + 1];

      v2f b0, b1, b2, b3;
      b0.x = W[(size_t)(0 * 16 + ncol) * 128 + kg + koff];
      b0.y = W[(size_t)(0 * 16 + ncol) * 128 + kg + koff + 1];
      b1.x = W[(size_t)(1 * 16 + ncol) * 128 + kg + koff];
      b1.y = W[(size_t)(1 * 16 + ncol) * 128 + kg + koff + 1];
      b2.x = W[(size_t)(2 * 16 + ncol) * 128 + kg + koff];
      b2.y = W[(size_t)(2 * 16 + ncol) * 128 + kg + koff + 1];
      b3.x = W[(size_t)(3 * 16 + ncol) * 128 + kg + koff];
      b3.y = W[(size_t)(3 * 16 + ncol) * 128 + kg + koff + 1];

      acc0 = __builtin_amdgcn_wmma_f32_16x16x4_f32(false, a, false, b0, (short)0, acc0, false, false);
      acc1 = __builtin_amdgcn_wmma_f32_16x16x4_f32(false, a, false, b1, (short)0, acc1, false, false);
      acc2 = __builtin_amdgcn_wmma_f32_16x16x4_f32(false, a, false, b2, (short)0, acc2, false, false);
      acc3 = __builtin_amdgcn_wmma_f32_16x16x4_f32(false, a, false, b3, (short)0, acc3, false, false);
    }
  }

  // Epilogue: C/D layout -> lanes 0-15 hold M = m_base+0..7 in VGPR 0..7 (N=lane),
  // lanes 16-31 hold M = m_base+8..15 (N=lane-16). Add bias, ReLU, store.
  const int mtop = m_base + ((lane >> 4) ? 8 : 0);
  float bv0 = bias[0 * 16 + ncol];
  float bv1 = bias[1 * 16 + ncol];
  float bv2 = bias[2 * 16 + ncol];
  float bv3 = bias[3 * 16 + ncol];
#pragma unroll
  for (int i = 0; i < 8; ++i) {
    float* orow = Uout + (size_t)(mtop + i) * EMB;
    float r0 = acc0[i] + bv0; orow[0 * 16 + ncol] = r0 > 0.f ? r0 : 0.f;
    float r1 = acc1[i] + bv1; orow[1 * 16 + ncol] = r1 > 0.f ? r1 : 0.f;
    float r2 = acc2[i] + bv2; orow[2 * 16 + ncol] = r2 > 0.f ? r2 : 0.f;
    float r3 = acc3[i] + bv3; orow[3 * 16 + ncol] = r3 > 0.f ? r3 : 0.f;
  }
}

// ---------------- launcher ----------------
extern "C" void kernel_launch(void* const* d_in, const int* in_sizes, int n_in,
                              void* d_out, int out_size, void* d_ws, size_t ws_size,
                              hipStream_t stream) {
  const float* user_emb = (const float*)d_in[0];  // [USER_NUM, 64]
  const float* item_emb = (const float*)d_in[1];  // [ITEM_NUM, 64]
  const float* W        = (const float*)d_in[2];  // [2, 64, 128]
  const float* b        = (const float*)d_in[3];  // [2, 64]
  const int*   s_rows   = (const int*)d_in[4];
  const int*   s_cols   = (const int*)d_in[5];
  const float* s_vals   = (const float*)d_in[6];
  const int*   r_rows   = (const int*)d_in[7];
  const int*   r_cols   = (const int*)d_in[8];
  const float* r_vals   = (const float*)d_in[9];

  float* out_user = (float*)d_out;                                // [USER_NUM*64]
  float* out_item = (float*)d_out + (size_t)USER_NUM * EMB;       // [ITEM_NUM*64]
  float* agg      = (float*)d_ws;                                 // [USER_NUM*64]

  const int nU = USER_NUM * EMB;          // 12.8M floats
  const dim3 blk(256);
  const dim3 zgrid((nU + 255) / 256);
  const dim3 sgrid((NE + 7) / 8);         // one wave per edge, 8 waves/block
  const dim3 dgrid((USER_NUM + 127) / 128);

  // ---- layer 0: agg = S @ user_emb ; U1 = relu([agg|U0] W0^T + b0) -> out_user
  zero_kernel<<<zgrid, blk, 0, stream>>>(agg, nU);
  spmm_atomic_kernel<<<sgrid, blk, 0, stream>>>(s_rows, s_cols, s_vals, user_emb, agg, NE);
  dense_layer_kernel<<<dgrid, blk, 0, stream>>>(agg, user_emb, W, b, out_user);

  // ---- layer 1: agg = S @ U1 ; U2 = relu([agg|U1] W1^T + b1) in-place on out_user
  zero_kernel<<<zgrid, blk, 0, stream>>>(agg, nU);
  spmm_atomic_kernel<<<sgrid, blk, 0, stream>>>(s_rows, s_cols, s_vals, out_user, agg, NE);
  dense_layer_kernel<<<dgrid, blk, 0, stream>>>(agg, out_user, W + 64 * 128, b + 64, out_user);

  // ---- user_all = U2 + R @ item_emb (atomic accumulate into out_user)
  spmm_atomic_kernel<<<sgrid, blk, 0, stream>>>(r_rows, r_cols, r_vals, item_emb, out_user, NE);

  // ---- item_all = item_emb
  const int n4 = (ITEM_NUM * EMB) / 4;
  copy4_kernel<<<(n4 + 255) / 256, blk, 0, stream>>>((const float4*)item_emb,
                                                     (float4*)out_item, n4);
}